// LearnableRouter_86131274154617
// MI455X (gfx1250) — compile-verified
//
#include <hip/hip_runtime.h>
#include <hip/hip_bf16.h>

typedef float v2f __attribute__((ext_vector_type(2)));
typedef float v8f __attribute__((ext_vector_type(8)));

#define B_   4
#define H_   16
#define L_   8192
#define D_   64
#define BS_  64
#define NQ_  128            // L_/BS_
#define SROW 129            // padded LDS row stride for scores (129 mod 64 = 1 -> conflict-light)

// ---------------------------------------------------------------------------
// Kernel 1: fused block-mean-pool + learned projection.
// grid = (B*H*NQ, 2)  (y=0 -> q/Wq, y=1 -> k/Wk), block = 64 threads.
// Each block handles one 64x64 f32 input tile: column sums via float4 loads,
// LDS reduce, then out[e] = (1/64) * sum_d colsum[d] * W[e][d].
// This is the bandwidth-dominant pass (268 MB streamed once, coalesced 16B).
// ---------------------------------------------------------------------------
__global__ __launch_bounds__(64)
void pool_project_kernel(const float* __restrict__ q, const float* __restrict__ k,
                         const float* __restrict__ Wq, const float* __restrict__ Wk,
                         float* __restrict__ qp, float* __restrict__ kp) {
    const float* x = blockIdx.y ? k : q;
    const float* W = blockIdx.y ? Wk : Wq;
    float* outp    = blockIdx.y ? kp : qp;

    const int bhblk = blockIdx.x;                 // bh*NQ + blk
    const int t  = threadIdx.x;                   // 0..63
    const int cg = t & 15;                        // column group of 4 floats
    const int rq = t >> 4;                        // row quarter (0..3)
    const size_t base = (size_t)bhblk * (BS_ * D_);   // tile start

    float4 ps = make_float4(0.f, 0.f, 0.f, 0.f);
#pragma unroll
    for (int r = 0; r < 16; ++r) {
        const int row = rq * 16 + r;
        float4 v = *(const float4*)(x + base + (size_t)row * D_ + cg * 4);
        ps.x += v.x; ps.y += v.y; ps.z += v.z; ps.w += v.w;
    }

    __shared__ float4 part[4][16];
    __shared__ float  colsum[D_];
    part[rq][cg] = ps;
    __syncthreads();
    if (rq == 0) {
        float4 a = part[0][cg], b = part[1][cg], c = part[2][cg], d = part[3][cg];
        colsum[cg * 4 + 0] = a.x + b.x + c.x + d.x;
        colsum[cg * 4 + 1] = a.y + b.y + c.y + d.y;
        colsum[cg * 4 + 2] = a.z + b.z + c.z + d.z;
        colsum[cg * 4 + 3] = a.w + b.w + c.w + d.w;
    }
    __syncthreads();

    // projection: thread t computes output feature e = t  (y = mean @ W^T)
    float acc = 0.f;
#pragma unroll
    for (int i = 0; i < 16; ++i) {
        float4 w = *(const float4*)(W + (size_t)t * D_ + i * 4);
        acc += w.x * colsum[i * 4 + 0] + w.y * colsum[i * 4 + 1] +
               w.z * colsum[i * 4 + 2] + w.w * colsum[i * 4 + 3];
    }
    outp[(size_t)bhblk * D_ + t] = acc * (1.0f / (float)BS_);
}

// ---------------------------------------------------------------------------
// Kernel 2: per-(b,h) scores via V_WMMA_F32_16X16X4_F32 + bias + soft_topk.
// grid = B*H = 64 blocks, 256 threads (8 wave32s).
// Wave w computes output rows [16w,16w+16). Both fragment sets are staged
// into register arrays (afrag once per strip, bfrag once per N-tile) so the
// 16-step K-accumulation issues as back-to-back v_wmma with no interleaved
// LDS waits (round-1 asm showed a serialized ds_load->wait->wmma chain).
// ---------------------------------------------------------------------------
__global__ __launch_bounds__(256)
void score_softtopk_kernel(const float* __restrict__ qp, const float* __restrict__ kp,
                           const float* __restrict__ bias,
                           const float* __restrict__ bias_scale_p,
                           float* __restrict__ out) {
    extern __shared__ float lds[];   // max(2*128*64, 128*SROW) floats

    const int bh  = blockIdx.x;
    const int h   = bh & (H_ - 1);
    const int tid = threadIdx.x;

    // ---- Phase A: stage q_proj / k_proj tiles (128x64 f32 each) into LDS ----
    {
        const float4* qs = (const float4*)(qp + (size_t)bh * NQ_ * D_);
        const float4* ks = (const float4*)(kp + (size_t)bh * NQ_ * D_);
        float4* lq = (float4*)lds;
        float4* lk = (float4*)(lds + NQ_ * D_);
#pragma unroll
        for (int i = tid; i < (NQ_ * D_) / 4; i += 256) {
            lq[i] = qs[i];
            lk[i] = ks[i];
        }
    }
    __syncthreads();

    const int wave = tid >> 5;
    const int lane = tid & 31;
    const int m0   = wave * 16;
    const int lm   = lane & 15;
    const int kh   = (lane >> 4) << 1;    // 0 or 2: K-pair selector per lane half

    // ---- Phase B: 16x128 strip of scores per wave via f32 WMMA ----
    // A fragments: lane's 64-float K-row of q_proj, resident in 32 VGPRs.
    v2f afrag[16];
#pragma unroll
    for (int kk = 0; kk < 16; ++kk)
        afrag[kk] = *(const v2f*)&lds[(m0 + lm) * D_ + kk * 4 + kh];

    v8f acc[8];
#pragma unroll
    for (int nt = 0; nt < 8; ++nt) {
        // Stage all 16 B fragments for this N-tile, then fire the WMMA chain.
        v2f bfrag[16];
#pragma unroll
        for (int kk = 0; kk < 16; ++kk)
            bfrag[kk] = *(const v2f*)&lds[NQ_ * D_ + (nt * 16 + lm) * D_ + kk * 4 + kh];

        v8f c = {0.f, 0.f, 0.f, 0.f, 0.f, 0.f, 0.f, 0.f};
#pragma unroll
        for (int kk = 0; kk < 16; ++kk)
            c = __builtin_amdgcn_wmma_f32_16x16x4_f32(
                    false, afrag[kk], false, bfrag[kk], (short)0, c, false, false);
        acc[nt] = c;
    }
    __syncthreads();   // all LDS tile reads complete before reuse

    // ---- write scaled scores ( (qk/sqrt(D) + bs*bias) / tau ) into LDS ----
    {
        const float bscale   = bias_scale_p[0];
        const float inv_sqd  = 0.125f;   // 1/sqrt(64)
        const float inv_tau  = 10.0f;    // 1/0.1
        const int mhalf = (lane >> 4) << 3;   // lanes 16-31 hold rows +8
#pragma unroll
        for (int nt = 0; nt < 8; ++nt) {
#pragma unroll
            for (int r = 0; r < 8; ++r) {
                const int m = m0 + r + mhalf;
                const int n = nt * 16 + lm;
                float v = acc[nt][r] * inv_sqd
                        + bscale * bias[((size_t)h * NQ_ + m) * NQ_ + n];
                lds[m * SROW + n] = v * inv_tau;
            }
        }
    }
    __syncthreads();

    // ---- Phase C: soft top-k bisection; 2 threads per row, half-row in regs ----
    const int row  = tid >> 1;
    const int half = tid & 1;
    float s[64];
#pragma unroll
    for (int j = 0; j < 64; ++j) s[j] = lds[row * SROW + half * 64 + j];

    float lo = -10000.0f, hi = 10000.0f;
    const float target = 0.15f * (float)NQ_;   // 19.2
    for (int it = 0; it < 20; ++it) {
        const float mid = 0.5f * (lo + hi);
        float p = 0.f;
#pragma unroll
        for (int j = 0; j < 64; ++j)
            p += 1.0f / (1.0f + __expf(-(s[j] + mid)));
        const float total = p + __shfl_xor(p, 1, 32);   // combine the two half-rows
        if (total < target) lo = mid; else hi = mid;
    }
    const float lam = 0.5f * (lo + hi);

    float* orow = out + ((size_t)bh * NQ_ + row) * NQ_ + half * 64;
#pragma unroll
    for (int j = 0; j < 64; ++j)
        orow[j] = 1.0f / (1.0f + __expf(-(s[j] + lam)));
}

// ---------------------------------------------------------------------------
extern "C" void kernel_launch(void* const* d_in, const int* in_sizes, int n_in,
                              void* d_out, int out_size, void* d_ws, size_t ws_size,
                              hipStream_t stream) {
    const float* q     = (const float*)d_in[0];
    const float* k     = (const float*)d_in[1];
    const float* bias  = (const float*)d_in[2];
    const float* Wq    = (const float*)d_in[3];
    const float* Wk    = (const float*)d_in[4];
    const float* bsc   = (const float*)d_in[5];
    float* out = (float*)d_out;

    float* qp = (float*)d_ws;                        // B*H*NQ*D f32 = 2 MB
    float* kp = qp + (size_t)B_ * H_ * NQ_ * D_;     // +2 MB

    dim3 g1(B_ * H_ * NQ_, 2);
    pool_project_kernel<<<g1, 64, 0, stream>>>(q, k, Wq, Wk, qp, kp);

    const size_t lds_bytes = (size_t)NQ_ * SROW * sizeof(float);  // 66 KB (> 2*128*64*4)
    score_softtopk_kernel<<<B_ * H_, 256, lds_bytes, stream>>>(qp, kp, bias, bsc, out);
}